// GNN_51539607552576
// MI455X (gfx1250) — compile-verified
//
#include <hip/hip_runtime.h>
#include <cmath>

typedef __attribute__((ext_vector_type(2))) float v2f;
typedef __attribute__((ext_vector_type(8))) float v8f;

#define HID   10
#define IND   256
#define NB    4
#define HWSZ  16384          // 128*128
#define NPIX  (NB*HWSZ)      // 65536 pixels

__device__ __forceinline__ float sigm(float x) { return 1.0f / (1.0f + __expf(-x)); }

// attention-gated fused update: x,m vec10; Wa [20]; Wu [10][20]
__device__ __forceinline__ void att_update(const float x[HID], const float m[HID],
                                           const float* __restrict__ Wa, float ba,
                                           const float* __restrict__ Wu,
                                           float o[HID]) {
  float cat[2*HID];
#pragma unroll
  for (int c = 0; c < HID; ++c) { cat[c] = x[c]; cat[HID+c] = m[c]; }
  float d = ba;
#pragma unroll
  for (int c = 0; c < 2*HID; ++c) d += Wa[c]*cat[c];
  float a = sigm(d);
#pragma unroll
  for (int oc = 0; oc < HID; ++oc) {
    float u = 0.f;
#pragma unroll
    for (int c = 0; c < 2*HID; ++c) u += Wu[oc*2*HID + c]*cat[c];
    u = fmaxf(u, 0.f);
    o[oc] = x[oc]*(1.f - a) + u*a;
  }
}

__global__ __launch_bounds__(256) void gnn_kernel(
    const float* __restrict__ xp, const float* __restrict__ xh, const float* __restrict__ xf,
    const float* __restrict__ p_fea, const float* __restrict__ h_fea,
    const float* __restrict__ Wf_comp, const float* __restrict__ bf_comp,
    const float* __restrict__ Wf_uatt, const float* __restrict__ bf_uatt, const float* __restrict__ Wf_uupd,
    const float* __restrict__ Wh_pdp_f, const float* __restrict__ Wh_pdp_x, const float* __restrict__ bh_pdp,
    const float* __restrict__ Wh_att,  const float* __restrict__ bh_att,
    const float* __restrict__ Wh_cu,   const float* __restrict__ bh_cu,
    const float* __restrict__ Wh_cl,   const float* __restrict__ bh_cl,
    const float* __restrict__ Wh_dec_f,const float* __restrict__ Wh_dec_x, const float* __restrict__ bh_dec,
    const float* __restrict__ Wh_uatt, const float* __restrict__ bh_uatt,  const float* __restrict__ Wh_uupd,
    const float* __restrict__ Wp_pdp_f,const float* __restrict__ Wp_pdp_x, const float* __restrict__ bp_pdp,
    const float* __restrict__ Wp_att,  const float* __restrict__ bp_att,
    const float* __restrict__ Wp_dec_f,const float* __restrict__ Wp_dec_x, const float* __restrict__ bp_dec,
    const float* __restrict__ Wp_uatt, const float* __restrict__ bp_uatt,  const float* __restrict__ Wp_uupd,
    float* __restrict__ out)
{
  // LDS: combined fp32 weight matrices for the two 256->16 projections, + WMMA results
  __shared__ float WF[2][IND][16];   // [set][k][n]  32 KB
  __shared__ float PROJ[2][128][16]; // [set][pixel-in-block][n]  16 KB

  const int tid = threadIdx.x;

  // ---- stage combined weights: set0 = h-graph (Wh_pdp_f | Wh_dec_f | 0), set1 = p-graph (Wp_pdp_f | Wp_dec_f)
  for (int e = tid; e < 2*IND*16; e += 256) {
    int s = e >> 12;          // /4096
    int k = (e >> 4) & 255;
    int n = e & 15;
    float v = 0.f;
    if (s == 0) {
      if (n < HID)        v = Wh_pdp_f[n*IND + k];
      else if (n < 12)    v = Wh_dec_f[(n-10)*IND + k];
    } else {
      if (n < HID)        v = Wp_pdp_f[n*IND + k];
      else                v = Wp_dec_f[(n-10)*IND + k];
    }
    WF[s][k][n] = v;
  }
  __syncthreads();

  // ---- Phase A: WMMA projections, one 16-pixel tile per wave (8 waves/block -> 128 pixels/block)
  {
    const int wave = tid >> 5;
    const int lane = tid & 31;
    const int laneHalf = lane >> 4;       // 0: K pair {0,1}; 1: K pair {2,3}
    const int laneIn   = lane & 15;       // M row (for A) / N col (for B)
    const int pix  = blockIdx.x*128 + wave*16 + laneIn;
    const int b    = pix >> 14;           // /16384
    const int hw   = pix & 16383;
    const float* hb = h_fea + (size_t)b*IND*HWSZ + hw;
    const float* pb = p_fea + (size_t)b*IND*HWSZ + hw;

    v8f accH = {}; v8f accP = {};
    for (int k0 = 0; k0 < IND; k0 += 4) {
      const int ka = k0 + laneHalf*2;
      v2f aH, aP, bH, bP;
      aH.x = hb[(size_t)ka*HWSZ];      aH.y = hb[(size_t)(ka+1)*HWSZ];
      aP.x = pb[(size_t)ka*HWSZ];      aP.y = pb[(size_t)(ka+1)*HWSZ];
      bH.x = WF[0][ka][laneIn];        bH.y = WF[0][ka+1][laneIn];
      bP.x = WF[1][ka][laneIn];        bP.y = WF[1][ka+1][laneIn];
      accH = __builtin_amdgcn_wmma_f32_16x16x4_f32(false, aH, false, bH, (short)0, accH, false, false);
      accP = __builtin_amdgcn_wmma_f32_16x16x4_f32(false, aP, false, bP, (short)0, accP, false, false);
    }
    // C/D layout: VGPR r holds M = r + 8*laneHalf, N = laneIn
    const int rowOff = wave*16 + laneHalf*8;
#pragma unroll
    for (int r = 0; r < 8; ++r) {
      PROJ[0][rowOff + r][laneIn] = accH[r];
      PROJ[1][rowOff + r][laneIn] = accP[r];
    }
  }
  __syncthreads();

  // ---- Phase B: per-pixel scalar graph update (threads 0..127, one pixel each)
  if (tid < 128) {
    const int p  = blockIdx.x*128 + tid;
    const int b  = p >> 14;
    const int hw = p & 16383;

    float Xp[6][HID], Xh[2][HID], Xf[HID];
#pragma unroll
    for (int n = 0; n < 6; ++n)
#pragma unroll
      for (int c = 0; c < HID; ++c)
        Xp[n][c] = xp[(((size_t)n*NB + b)*HID + c)*HWSZ + hw];
#pragma unroll
    for (int n = 0; n < 2; ++n)
#pragma unroll
      for (int c = 0; c < HID; ++c)
        Xh[n][c] = xh[(((size_t)n*NB + b)*HID + c)*HWSZ + hw];
#pragma unroll
    for (int c = 0; c < HID; ++c)
      Xf[c] = xf[((size_t)b*HID + c)*HWSZ + hw];

    float hprojv[12], pprojv[16];
#pragma unroll
    for (int i = 0; i < 12; ++i) hprojv[i] = PROJ[0][tid][i];
#pragma unroll
    for (int i = 0; i < 16; ++i) pprojv[i] = PROJ[1][tid][i];

    // ================= Full graph =================
    float xf_msg[HID];
#pragma unroll
    for (int c = 0; c < HID; ++c) xf_msg[c] = 0.f;
#pragma unroll
    for (int j = 0; j < 2; ++j) {
      float d = bf_comp[j];
#pragma unroll
      for (int c = 0; c < HID; ++c) d += Wf_comp[j*HID + c]*Xh[j][c];
      float a = sigm(d);
#pragma unroll
      for (int c = 0; c < HID; ++c) xf_msg[c] += a*Xh[j][c];
    }
    float xf_new[HID];
    att_update(Xf, xf_msg, Wf_uatt, bf_uatt[0], Wf_uupd, xf_new);

    // ================= Half graph =================
    float hatt[2], hdec[2];
#pragma unroll
    for (int i = 0; i < 2; ++i) {
      float d = bh_att[i];
#pragma unroll
      for (int c = 0; c < HID; ++c) d += Wh_att[i*HID + c]*Xh[i][c];
      hatt[i] = sigm(d);
      float dd = hprojv[10 + i] + bh_dec[i];
#pragma unroll
      for (int c = 0; c < HID; ++c) dd += Wh_dec_x[i*HID + c]*Xh[i][c];
      hdec[i] = sigm(dd);
    }
    float hdp[2][HID];
#pragma unroll
    for (int i = 0; i < 2; ++i)
#pragma unroll
      for (int o = 0; o < HID; ++o) {
        float d = hprojv[o] + bh_pdp[o];
#pragma unroll
        for (int c = 0; c < HID; ++c) d += Wh_pdp_x[o*HID + c]*Xh[i][c];
        hdp[i][o] = fmaxf(d, 0.f);
      }
    float xphu[HID], xphl[HID];
#pragma unroll
    for (int c = 0; c < HID; ++c) { xphu[c] = 0.f; xphl[c] = 0.f; }
#pragma unroll
    for (int j = 0; j < 4; ++j) {
      float d = bh_cu[j];
#pragma unroll
      for (int c = 0; c < HID; ++c) d += Wh_cu[j*HID + c]*Xp[j][c];
      float a = sigm(d);
#pragma unroll
      for (int c = 0; c < HID; ++c) xphu[c] += a*Xp[j][c];
    }
#pragma unroll
    for (int j = 0; j < 2; ++j) {
      float d = bh_cl[j];
#pragma unroll
      for (int c = 0; c < HID; ++c) d += Wh_cl[j*HID + c]*Xp[4+j][c];
      float a = sigm(d);
#pragma unroll
      for (int c = 0; c < HID; ++c) xphl[c] += a*Xp[4+j][c];
    }
    float xh_new[2][HID];
#pragma unroll
    for (int i = 0; i < 2; ++i) {
      float msg[HID];
      const float gx = (i == 0) ? hatt[1]*(1.f - hatt[0]) : hatt[0]*(1.f - hatt[1]);
      const int other = 1 - i;
#pragma unroll
      for (int c = 0; c < HID; ++c)
        msg[c] = ((i == 0) ? xphu[c] : xphl[c]) + hdp[other][c]*gx + hdec[i]*Xf[c];
      att_update(Xh[i], msg, Wh_uatt + i*2*HID, bh_uatt[i], Wh_uupd + i*HID*2*HID, xh_new[i]);
    }

    // ================= Part graph =================
    float patt[6];
#pragma unroll
    for (int i = 0; i < 6; ++i) {
      float d = bp_att[i];
#pragma unroll
      for (int c = 0; c < HID; ++c) d += Wp_att[i*HID + c]*Xp[i][c];
      patt[i] = sigm(d);
    }
    float pdp[6][HID];
#pragma unroll
    for (int i = 0; i < 6; ++i)
#pragma unroll
      for (int o = 0; o < HID; ++o) {
        float d = pprojv[o] + bp_pdp[o];
#pragma unroll
        for (int c = 0; c < HID; ++c) d += Wp_pdp_x[o*HID + c]*Xp[i][c];
        pdp[i][o] = fmaxf(d, 0.f);
      }
    float xpp[6][HID];
#pragma unroll
    for (int i = 0; i < 6; ++i)
#pragma unroll
      for (int c = 0; c < HID; ++c) xpp[i][c] = 0.f;
    // edges (src,dst) from argwhere(ADJ), row-major
    const int esrc[12] = {0,1,1,1,1,1,2,3,4,4,5,5};
    const int edst[12] = {1,0,2,3,4,5,1,1,1,5,1,4};
#pragma unroll
    for (int e = 0; e < 12; ++e) {
      const int s = esrc[e], d2 = edst[e];
      const float g = patt[s]*(1.f - patt[d2]);
#pragma unroll
      for (int c = 0; c < HID; ++c) xpp[d2][c] += pdp[s][c]*g;
    }
    // decomposition gate + update, write xp_new directly
#pragma unroll
    for (int i = 0; i < 6; ++i) {
      float d = pprojv[10 + i] + bp_dec[i];
#pragma unroll
      for (int c = 0; c < HID; ++c) d += Wp_dec_x[i*HID + c]*Xp[i][c];
      const float a = sigm(d);
      const int hi = (i < 4) ? 0 : 1;   // HALF_IDX
      float msg[HID];
#pragma unroll
      for (int c = 0; c < HID; ++c) msg[c] = xpp[i][c] + a*Xh[hi][c];
      float xpn[HID];
      att_update(Xp[i], msg, Wp_uatt + i*2*HID, bp_uatt[i], Wp_uupd + i*HID*2*HID, xpn);
#pragma unroll
      for (int c = 0; c < HID; ++c)
        out[(((size_t)i*NB + b)*HID + c)*HWSZ + hw] = xpn[c];
    }
    // halves -> nodes 6,7 ; full -> node 8
#pragma unroll
    for (int i = 0; i < 2; ++i)
#pragma unroll
      for (int c = 0; c < HID; ++c)
        out[(((size_t)(6 + i)*NB + b)*HID + c)*HWSZ + hw] = xh_new[i][c];
#pragma unroll
    for (int c = 0; c < HID; ++c)
      out[(((size_t)8*NB + b)*HID + c)*HWSZ + hw] = xf_new[c];
  }
}

extern "C" void kernel_launch(void* const* d_in, const int* in_sizes, int n_in,
                              void* d_out, int out_size, void* d_ws, size_t ws_size,
                              hipStream_t stream) {
  (void)in_sizes; (void)n_in; (void)d_ws; (void)ws_size; (void)out_size;
  const float* xp       = (const float*)d_in[0];
  const float* xh       = (const float*)d_in[1];
  const float* xf       = (const float*)d_in[2];
  // d_in[3] = bg_node (unused by reference)
  const float* p_fea    = (const float*)d_in[4];
  const float* h_fea    = (const float*)d_in[5];
  // d_in[6] = f_fea (unused by reference)
  const float* Wf_comp  = (const float*)d_in[7];
  const float* bf_comp  = (const float*)d_in[8];
  const float* Wf_uatt  = (const float*)d_in[9];
  const float* bf_uatt  = (const float*)d_in[10];
  const float* Wf_uupd  = (const float*)d_in[11];
  const float* Wh_pdp_f = (const float*)d_in[12];
  const float* Wh_pdp_x = (const float*)d_in[13];
  const float* bh_pdp   = (const float*)d_in[14];
  const float* Wh_att   = (const float*)d_in[15];
  const float* bh_att   = (const float*)d_in[16];
  const float* Wh_cu    = (const float*)d_in[17];
  const float* bh_cu    = (const float*)d_in[18];
  const float* Wh_cl    = (const float*)d_in[19];
  const float* bh_cl    = (const float*)d_in[20];
  const float* Wh_dec_f = (const float*)d_in[21];
  const float* Wh_dec_x = (const float*)d_in[22];
  const float* bh_dec   = (const float*)d_in[23];
  const float* Wh_uatt  = (const float*)d_in[24];
  const float* bh_uatt  = (const float*)d_in[25];
  const float* Wh_uupd  = (const float*)d_in[26];
  const float* Wp_pdp_f = (const float*)d_in[27];
  const float* Wp_pdp_x = (const float*)d_in[28];
  const float* bp_pdp   = (const float*)d_in[29];
  const float* Wp_att   = (const float*)d_in[30];
  const float* bp_att   = (const float*)d_in[31];
  const float* Wp_dec_f = (const float*)d_in[32];
  const float* Wp_dec_x = (const float*)d_in[33];
  const float* bp_dec   = (const float*)d_in[34];
  const float* Wp_uatt  = (const float*)d_in[35];
  const float* bp_uatt  = (const float*)d_in[36];
  const float* Wp_uupd  = (const float*)d_in[37];

  float* out = (float*)d_out;
  dim3 grid(NPIX / 128);   // 512 blocks, 128 pixels each
  dim3 block(256);
  gnn_kernel<<<grid, block, 0, stream>>>(
      xp, xh, xf, p_fea, h_fea,
      Wf_comp, bf_comp, Wf_uatt, bf_uatt, Wf_uupd,
      Wh_pdp_f, Wh_pdp_x, bh_pdp, Wh_att, bh_att,
      Wh_cu, bh_cu, Wh_cl, bh_cl,
      Wh_dec_f, Wh_dec_x, bh_dec, Wh_uatt, bh_uatt, Wh_uupd,
      Wp_pdp_f, Wp_pdp_x, bp_pdp, Wp_att, bp_att,
      Wp_dec_f, Wp_dec_x, bp_dec, Wp_uatt, bp_uatt, Wp_uupd,
      out);
}